// MoEBlock_49924699848918
// MI455X (gfx1250) — compile-verified
//
#include <hip/hip_runtime.h>
#include <math.h>

#define H    512
#define H2   1024
#define NE   8
#define MT   64          // tokens per block tile
#define XS   516         // x LDS stride   (516 % 64 == 4 -> conflict-free columns)
#define CS   132         // h-chunk stride (132 % 64 == 4)
#define NCH  128         // h columns per chunk

typedef __attribute__((ext_vector_type(2))) float v2f;
typedef __attribute__((ext_vector_type(8))) float v8f;

// ---------------------------------------------------------------------------
// Zero output (incl. trailing aux_loss scalar) and per-expert counters.
// ---------------------------------------------------------------------------
__global__ void moe_zero_kernel(float* __restrict__ out, int n,
                                unsigned* __restrict__ counts) {
    int gid = blockIdx.x * blockDim.x + threadIdx.x;
    if (gid < NE) counts[gid] = 0u;
    for (int i = gid; i < n; i += gridDim.x * blockDim.x) out[i] = 0.0f;
}

// ---------------------------------------------------------------------------
// Router: logits = x @ Wr, top-2, renormalize, scatter (token, weight) into
// per-expert lists. Renormalized top-2 softmax == 2-way softmax of top logits.
// ---------------------------------------------------------------------------
__global__ __launch_bounds__(256)
void moe_router_kernel(const float* __restrict__ x,
                       const float* __restrict__ Wr,
                       unsigned* __restrict__ counts,
                       int* __restrict__ tok_list,
                       float* __restrict__ w_list,
                       int T) {
    __shared__ float wrs[H * NE];           // 16 KB, row-major [H][E]
    int tid = threadIdx.x;
    for (int i = tid; i < H * NE; i += blockDim.x) wrs[i] = Wr[i];
    __syncthreads();

    int t = blockIdx.x * blockDim.x + tid;
    if (t >= T) return;

    const float4* xr = (const float4*)(x + (size_t)t * H);
    float l[NE];
#pragma unroll
    for (int e = 0; e < NE; ++e) l[e] = 0.0f;
    for (int k4 = 0; k4 < H / 4; ++k4) {
        float4 v = xr[k4];
        const float* w = &wrs[(k4 * 4) * NE];
#pragma unroll
        for (int e = 0; e < NE; ++e)
            l[e] += v.x * w[e] + v.y * w[NE + e] + v.z * w[2 * NE + e] + v.w * w[3 * NE + e];
    }
    // top-2 with lowest-index tie-breaking (matches lax.top_k)
    int i0 = 0; float b0 = l[0];
#pragma unroll
    for (int e = 1; e < NE; ++e) if (l[e] > b0) { b0 = l[e]; i0 = e; }
    int i1 = (i0 == 0) ? 1 : 0; float bv = l[i1];
#pragma unroll
    for (int e = 0; e < NE; ++e) if (e != i0 && l[e] > bv) { bv = l[e]; i1 = e; }

    float w0 = 1.0f / (1.0f + expf(bv - b0));   // renormalized pair weights
    float w1 = 1.0f - w0;

    unsigned p = atomicAdd(&counts[i0], 1u);
    tok_list[(size_t)i0 * T + p] = t;  w_list[(size_t)i0 * T + p] = w0;
    p = atomicAdd(&counts[i1], 1u);
    tok_list[(size_t)i1 * T + p] = t;  w_list[(size_t)i1 * T + p] = w1;
}

// ---------------------------------------------------------------------------
// Fused expert FFN over one expert's token list, 64-token tiles per block.
// 256 threads = 8 waves; fp32 WMMA 16x16x4 throughout.
// h is produced/consumed in 128-column chunks through LDS; the [64 x 512]
// output tile is held in persistent accumulators (16 v8f per wave).
// Each B fragment feeds 4 WMMAs (4 M-tiles) -> 0.5 global loads per WMMA.
// ---------------------------------------------------------------------------
__global__ __launch_bounds__(256)
void moe_ffn_kernel(const float* __restrict__ x,
                    const float* __restrict__ W1,
                    const float* __restrict__ b1,
                    const float* __restrict__ W2,
                    const float* __restrict__ b2,
                    const unsigned* __restrict__ counts,
                    const int* __restrict__ tok_list,
                    const float* __restrict__ w_list,
                    float* __restrict__ out,
                    int e, int T) {
    __shared__ __align__(16) float xs[MT * XS];   // 132.1 KB gathered x tile
    __shared__ __align__(16) float hc[MT * CS];   //  33.8 KB gelu(h) chunk
    __shared__ int   tok_s[MT];
    __shared__ float w_s[MT];

    int cnt = (int)counts[e];
    int tile = blockIdx.x;
    if (tile * MT >= cnt) return;                 // uniform exit

    const float* W1e = W1 + (size_t)e * H * H2;
    const float* b1e = b1 + (size_t)e * H2;
    const float* W2e = W2 + (size_t)e * H2 * H;
    const float* b2e = b2 + (size_t)e * H;

    int tid  = threadIdx.x;
    int wv   = tid >> 5;
    int lane = tid & 31;
    int m16  = lane & 15;
    int half = lane >> 4;

    if (tid < MT) {
        int idx = tile * MT + tid;
        bool vld = idx < cnt;
        tok_s[tid] = vld ? tok_list[(size_t)e * T + idx] : 0;
        w_s[tid]   = vld ? w_list[(size_t)e * T + idx] : 0.0f;
    }
    __syncthreads();

    // gather 64 token rows of x into LDS (float4, coalesced per row)
#pragma unroll 4
    for (int j = 0; j < (MT * H / 4) / 256; ++j) {        // 32 iterations
        int idx = tid + 256 * j;
        int row = idx >> 7, c4 = idx & 127;
        const float4* src = (const float4*)(x + (size_t)tok_s[row] * H);
        *(float4*)&xs[row * XS + c4 * 4] = src[c4];
    }
    __syncthreads();

    int nb2 = wv * 64;                            // this wave's output columns
    v8f acc2[4][4] = {};                          // [nt][mt], persistent

    for (int c = 0; c < H2 / NCH; ++c) {          // 8 chunks of 128 h-columns
        int C0  = c * NCH;
        int n0w = C0 + wv * 16;                   // wave's 16 h-cols this chunk

        // ---- stage 1 partial: h[:, n0w..n0w+15] = gelu(x @ W1 + b1) -------
        v8f acc1[4] = {};
        for (int k0 = 0; k0 < H; k0 += 4) {
            int kk = k0 + 2 * half;               // A/B frag K row (ISA layout)
            const float* w1row = W1e + (size_t)kk * H2 + n0w + m16;
            v2f b; b.x = w1row[0]; b.y = w1row[H2];
#pragma unroll
            for (int mt = 0; mt < 4; ++mt) {
                v2f a = *(const v2f*)&xs[(mt * 16 + m16) * XS + kk];
                acc1[mt] = __builtin_amdgcn_wmma_f32_16x16x4_f32(
                    false, a, false, b, (short)0, acc1[mt], false, false);
            }
        }
        {
            int nl = wv * 16 + m16;               // local column in chunk
            float bb = b1e[C0 + nl];
#pragma unroll
            for (int mt = 0; mt < 4; ++mt)
#pragma unroll
                for (int r = 0; r < 8; ++r) {     // C layout: M = r + 8*half
                    float hv = acc1[mt][r] + bb;
                    hv = 0.5f * hv * (1.0f + erff(hv * 0.70710678118654752440f));
                    hc[(mt * 16 + r + 8 * half) * CS + nl] = hv;
                }
        }
        __syncthreads();

        // ---- stage 2 partial: acc2 += h_chunk @ W2[C0:C0+128, :] ----------
        for (int kk0 = 0; kk0 < NCH; kk0 += 4) {
            int kl = kk0 + 2 * half;
            const float* w2row = W2e + (size_t)(C0 + kl) * H + nb2 + m16;
            v2f bf[4];
#pragma unroll
            for (int nt = 0; nt < 4; ++nt) {
                bf[nt].x = w2row[nt * 16];
                bf[nt].y = w2row[nt * 16 + H];
            }
#pragma unroll
            for (int mt = 0; mt < 4; ++mt) {
                v2f a = *(const v2f*)&hc[(mt * 16 + m16) * CS + kl];
#pragma unroll
                for (int nt = 0; nt < 4; ++nt)
                    acc2[nt][mt] = __builtin_amdgcn_wmma_f32_16x16x4_f32(
                        false, a, false, bf[nt], (short)0, acc2[nt][mt],
                        false, false);
            }
        }
        __syncthreads();                          // hc reused next chunk
    }

    // accumulate weighted output; each token touched once per expert launch,
    // launches serialized on the stream in ascending e -> deterministic.
    int valid = cnt - tile * MT; if (valid > MT) valid = MT;
#pragma unroll
    for (int nt = 0; nt < 4; ++nt) {
        int n = nb2 + nt * 16 + m16;
        float bb = b2e[n];
#pragma unroll
        for (int mt = 0; mt < 4; ++mt)
#pragma unroll
            for (int r = 0; r < 8; ++r) {
                int m = mt * 16 + r + 8 * half;
                if (m < valid) {
                    int t = tok_s[m];
                    float y = acc2[nt][mt][r] + bb;
                    out[(size_t)t * H + n] += w_s[m] * y;
                }
            }
    }
}

// ---------------------------------------------------------------------------
extern "C" void kernel_launch(void* const* d_in, const int* in_sizes, int n_in,
                              void* d_out, int out_size, void* d_ws, size_t ws_size,
                              hipStream_t stream) {
    const float* x  = (const float*)d_in[0];
    const float* Wr = (const float*)d_in[1];
    const float* W1 = (const float*)d_in[2];
    const float* b1 = (const float*)d_in[3];
    const float* W2 = (const float*)d_in[4];
    const float* b2 = (const float*)d_in[5];
    float* out = (float*)d_out;

    int T = in_sizes[0] / H;                      // 32768 tokens

    unsigned* counts = (unsigned*)d_ws;
    int*   tok_list  = (int*)((char*)d_ws + 64);
    float* w_list    = (float*)((char*)d_ws + 64 + (size_t)NE * T * sizeof(int));

    moe_zero_kernel<<<2048, 256, 0, stream>>>(out, out_size, counts);
    moe_router_kernel<<<(T + 255) / 256, 256, 0, stream>>>(x, Wr, counts,
                                                           tok_list, w_list, T);
    int maxTiles = (T + MT - 1) / MT;             // upper bound; blocks self-trim
    for (int e = 0; e < NE; ++e) {
        moe_ffn_kernel<<<maxTiles, 256, 0, stream>>>(x, W1, b1, W2, b2, counts,
                                                     tok_list, w_list, out, e, T);
    }
}